// SimpleRNN_39814346834218
// MI455X (gfx1250) — compile-verified
//
#include <hip/hip_runtime.h>

// CDNA5 / gfx1250 RNN:  h_{t+1} = tanh(W_hh h_t + x_t*W_ih + b),  out = sigmoid(W_fc h_T + b_fc)
// One wave = 32 batch rows.  D = bigA x B + C with V_WMMA_F32_16X16X4_F32, where bigA is a
// constant row/column-permuted block-diagonal embedding of W_hh chosen so that the D register
// layout of step t IS the B register layout of step t+1 (zero cross-lane shuffles per step).

typedef __attribute__((ext_vector_type(2))) float v2f;
typedef __attribute__((ext_vector_type(8))) float v8f;

#define RNN_T 1024
#define RNN_H 8

__device__ __forceinline__ float fast_tanh(float v) {
#if __has_builtin(__builtin_amdgcn_tanhf)
  return __builtin_amdgcn_tanhf(v);           // v_tanh_f32 (gfx1250 trans op)
#else
  // tanh(v) = 1 - 2/(exp(2v)+1); exp via v_exp_f32
  float e = __builtin_amdgcn_exp2f(v * 2.8853900817779268f); // 2*log2(e)
  return 1.0f - 2.0f * __builtin_amdgcn_rcpf(e + 1.0f);
#endif
}

__global__ __launch_bounds__(64, 1) void rnn_wmma_f32_kernel(
    const float* __restrict__ x,     // [B, T]
    const float* __restrict__ W_ih,  // [H, 1]
    const float* __restrict__ W_hh,  // [H, H]
    const float* __restrict__ b_ih,  // [H]
    const float* __restrict__ b_hh,  // [H]
    const float* __restrict__ W_fc,  // [1, H]
    const float* __restrict__ b_fc,  // [1]
    float* __restrict__ out)         // [B, 1]
{
  const int tid  = blockIdx.x * blockDim.x + threadIdx.x;   // == batch index
  const int lane = threadIdx.x & 31;

  // ---- uniform parameters (scalar loads, hoisted out of the loop) ----
  float wih[RNN_H], bsum[RNN_H], wfc[RNN_H];
#pragma unroll
  for (int v = 0; v < RNN_H; ++v) {
    wih[v]  = W_ih[v];
    bsum[v] = b_ih[v] + b_hh[v];
    wfc[v]  = W_fc[v];
  }
  const float bfc = b_fc[0];

  // ---- build constant bigA (16x16, two permuted copies of W_hh) in A-layout ----
  // A-layout (32-bit, 16x4 tile): lane 0-15 -> M=lane, lanes 16-31 -> M=lane-16,
  // VGPR c holds K=c (lanes lo) / K=c+2 (lanes hi) within the tile.
  // bigA[M][K] = W_hh[M&7][2*(K>>2)+(K&1)]  iff (M>>3)==((K>>1)&1), else 0.
  const int M   = lane & 15;
  const int khi = (lane >> 4) << 1;  // 0 for lanes 0-15, 2 for lanes 16-31
  v2f A[4];
#pragma unroll
  for (int t = 0; t < 4; ++t) {
    v2f a;
#pragma unroll
    for (int c = 0; c < 2; ++c) {
      const int K   = 4 * t + c + khi;
      const bool nz = ((M >> 3) == ((K >> 1) & 1));
      const int row = M & 7;
      const int col = 2 * (K >> 2) + (K & 1);
      a[c] = nz ? W_hh[row * RNN_H + col] : 0.0f;
    }
    A[t] = a;
  }

  // ---- recurrence: h components double as the four B tiles of the next step ----
  v8f h = {};  // h_0 = 0
  const float* xrow = x + (size_t)tid * RNN_T;
  const float4* x4  = (const float4*)xrow;

  float4 xv = x4[0];
  for (int tb = 0; tb < RNN_T / 4; ++tb) {
    int nb = tb + 1;
    nb = (nb < RNN_T / 4) ? nb : (RNN_T / 4 - 1);
    float4 xn = x4[nb];  // software prefetch: 4 steps of latency slack

#pragma unroll
    for (int s = 0; s < 4; ++s) {
      const float xs = (s == 0) ? xv.x : (s == 1) ? xv.y : (s == 2) ? xv.z : xv.w;

      // C = pre_t  (hidden v in VGPR v, batch = lane)
      v8f c;
#pragma unroll
      for (int v = 0; v < RNN_H; ++v) c[v] = fmaf(xs, wih[v], bsum[v]);

      // D = bigA * [h-tiles] + C ; four K=4 slices chained through the accumulator
      v2f b0 = {h[0], h[1]};
      v2f b1 = {h[2], h[3]};
      v2f b2 = {h[4], h[5]};
      v2f b3 = {h[6], h[7]};
      v8f d;
      d = __builtin_amdgcn_wmma_f32_16x16x4_f32(false, A[0], false, b0, (short)0, c, false, false);
      d = __builtin_amdgcn_wmma_f32_16x16x4_f32(false, A[1], false, b1, (short)0, d, false, false);
      d = __builtin_amdgcn_wmma_f32_16x16x4_f32(false, A[2], false, b2, (short)0, d, false, false);
      d = __builtin_amdgcn_wmma_f32_16x16x4_f32(false, A[3], false, b3, (short)0, d, false, false);

      // h = tanh(D); register layout already matches next step's B tiles
#pragma unroll
      for (int v = 0; v < RNN_H; ++v) h[v] = fast_tanh(d[v]);
    }
    xv = xn;
  }

  // ---- head: sigmoid(h . W_fc + b_fc), fully per-lane, coalesced store ----
  float z = bfc;
#pragma unroll
  for (int v = 0; v < RNN_H; ++v) z = fmaf(h[v], wfc[v], z);
  float e = __builtin_amdgcn_exp2f(-z * 1.4426950408889634f);
  out[tid] = __builtin_amdgcn_rcpf(1.0f + e);
}

extern "C" void kernel_launch(void* const* d_in, const int* in_sizes, int n_in,
                              void* d_out, int out_size, void* d_ws, size_t ws_size,
                              hipStream_t stream) {
  (void)n_in; (void)d_ws; (void)ws_size; (void)out_size;
  const float* x    = (const float*)d_in[0];
  const float* W_ih = (const float*)d_in[1];
  const float* W_hh = (const float*)d_in[2];
  const float* b_ih = (const float*)d_in[3];
  const float* b_hh = (const float*)d_in[4];
  const float* W_fc = (const float*)d_in[5];
  const float* b_fc = (const float*)d_in[6];
  float* out = (float*)d_out;

  const int B = in_sizes[0] / RNN_T;  // 8192
  dim3 block(64);                     // 2 waves per workgroup
  dim3 grid(B / 64);                  // 128 workgroups -> 256 waves, 32 batches/wave
  rnn_wmma_f32_kernel<<<grid, block, 0, stream>>>(x, W_ih, W_hh, b_ih, b_hh, W_fc, b_fc, out);
}